// PadWithin_3496103379336
// MI455X (gfx1250) — compile-verified
//
#include <hip/hip_runtime.h>
#include <cstdint>
#include <cstddef>

// Zero-insertion upsampling (stride 2), MI455X / gfx1250.
//
// out[b,c,2i,2j] = in[b,c,i,j]; all other outputs zero.
// Pure bandwidth problem: 64 MiB read + 256 MiB written-once.
// - even output rows: {a,0,b,0} built in VGPRs, non-temporal b128 stores
// - odd output rows (all zero): CDNA5 async LDS->global DMA
//   (global_store_async_from_lds_b128, ASYNCcnt) from a 16B zero block in LDS.

// True clang vector types: required by __builtin_nontemporal_load/store
// (HIP_vector_type structs are rejected by the builtin).
typedef float vf2 __attribute__((ext_vector_type(2)));
typedef float vf4 __attribute__((ext_vector_type(4)));

__global__ __launch_bounds__(256) void upsample2_fast_kernel(
    const float* __restrict__ in, const int* __restrict__ stride_p,
    float* __restrict__ out) {
  if (*stride_p != 2) return;  // uniform scalar branch

  // 16 bytes of zeros staged in LDS; every lane's async store reads it.
  __shared__ alignas(16) uint32_t zbuf[4];
  if (threadIdx.x == 0) {
    zbuf[0] = 0u; zbuf[1] = 0u; zbuf[2] = 0u; zbuf[3] = 0u;
  }
  __syncthreads();  // also restores full EXEC before the async store
  // Generic pointer to LDS: low 32 bits are the LDS byte offset.
  uint32_t lds_off = (uint32_t)(uintptr_t)(&zbuf[0]);

  // 131072 input rows of 128 floats; 64 threads per row, one float4 of
  // output each. 8,388,608 threads total, exact cover (no tail).
  size_t t = (size_t)blockIdx.x * blockDim.x + threadIdx.x;
  size_t r = t >> 6;   // input row   [0, 16*64*128)
  size_t c = t & 63;   // float4 slot [0, 64) within the 256-wide output row

  const vf2* __restrict__ in2 = (const vf2*)in;
  vf2 a = __builtin_nontemporal_load(&in2[r * 64 + c]);  // 8B coalesced

  vf4 v = {a.x, 0.0f, a.y, 0.0f};
  vf4* out4 = (vf4*)out;
  size_t data_idx = (r * 2) * 64 + c;               // even output row
  __builtin_nontemporal_store(v, &out4[data_idx]);  // b128 NT store

  // Odd output row (all zeros): async DMA 16B of zeros from LDS to global.
  uint64_t zaddr = (uint64_t)(uintptr_t)(&out4[data_idx + 64]);
  asm volatile("global_store_async_from_lds_b128 %0, %1, off"
               :
               : "v"(zaddr), "v"(lds_off)
               : "memory");
  // s_endpgm implicitly waits for all counters; explicit wait documents it.
  asm volatile("s_wait_asynccnt 0" ::: "memory");
}

// Fallback for stride != 2 (never taken for the reference setup, but keeps
// the launch correct for any stride). Grid-stride loop, one output element
// per iteration.
__global__ __launch_bounds__(256) void upsample_generic_kernel(
    const float* __restrict__ in, const int* __restrict__ stride_p,
    float* __restrict__ out, size_t total_out) {
  int s = *stride_p;
  if (s == 2) return;  // fast kernel handled it; cheap uniform exit
  const size_t H = 128, W = 128;
  size_t W_out = W * (size_t)s;
  size_t H_out = H * (size_t)s;
  size_t gs = (size_t)gridDim.x * blockDim.x;
  for (size_t o = (size_t)blockIdx.x * blockDim.x + threadIdx.x; o < total_out;
       o += gs) {
    size_t row = o / W_out;
    size_t j = o - row * W_out;
    size_t i = row % H_out;
    size_t bc = row / H_out;
    float v = 0.0f;
    if ((i % (size_t)s) == 0 && (j % (size_t)s) == 0) {
      v = in[(bc * H + i / (size_t)s) * W + j / (size_t)s];
    }
    out[o] = v;
  }
}

extern "C" void kernel_launch(void* const* d_in, const int* in_sizes, int n_in,
                              void* d_out, int out_size, void* d_ws,
                              size_t ws_size, hipStream_t stream) {
  const float* feats = (const float*)d_in[0];
  const int* stride_p = (const int*)d_in[1];
  float* out = (float*)d_out;

  // Fast path: 16*64*128 rows * 64 float4-chunks = 8,388,608 threads.
  dim3 block(256);
  dim3 grid_fast(32768);
  upsample2_fast_kernel<<<grid_fast, block, 0, stream>>>(feats, stride_p, out);

  // Generic fallback (early-exits when stride==2); small fixed grid so the
  // wasted launch is negligible.
  dim3 grid_gen(4096);
  upsample_generic_kernel<<<grid_gen, block, 0, stream>>>(
      feats, stride_p, out, (size_t)out_size);
}